// AdjacencyAttention_58308476010999
// MI455X (gfx1250) — compile-verified
//
#include <hip/hip_runtime.h>
#include <math.h>

typedef __attribute__((ext_vector_type(2))) float v2f;
typedef __attribute__((ext_vector_type(8))) float v8f;

#define NN 64
#define DD 256
#define PITCH 68   // 64 + 4: keeps 16B row alignment, breaks bank-64 alignment

__global__ __launch_bounds__(256) void adj_attn_kernel(
    const float* __restrict__ x,
    const float* __restrict__ W1,
    const float* __restrict__ W2,
    const float* __restrict__ W3,
    const float* __restrict__ bs,
    const float* __restrict__ Vs,
    float* __restrict__ out)
{
    __shared__ float sW2[DD];
    __shared__ float sW3[DD];
    __shared__ float sVs[NN * PITCH];
    __shared__ float sT [NN * PITCH];   // t matrix, then reused for scores s
    __shared__ float sLhs[NN];
    __shared__ float sRhs[NN];
    __shared__ float sPart[NN * 8];
    __shared__ float sRed[8];

    const int tid  = threadIdx.x;
    const int lane = tid & 31;
    const int wave = tid >> 5;
    const int b    = blockIdx.x;

    // ---- stage W2/W3 and Vs into LDS ----
    if (tid < DD) { sW2[tid] = W2[tid]; sW3[tid] = W3[tid]; }
#pragma unroll
    for (int i = 0; i < 16; ++i) {
        int idx = tid + i * 256;                 // 0..4095
        sVs[(idx >> 6) * PITCH + (idx & 63)] = Vs[idx];
    }
    const float w1 = W1[0];
    __syncthreads();

    // ---- phase 1: lhs = (x W2)*W1, rhs = x W3 (stream x once) ----
    {
        const int row = tid >> 2;                // 0..63
        const int q   = tid & 3;                 // quarter of D
        const float4* xr = (const float4*)(x + ((size_t)b * NN + row) * DD + q * 64);
        float pl = 0.f, pr = 0.f;
#pragma unroll
        for (int i = 0; i < 16; ++i) {
            float4 v = xr[i];
            int d = q * 64 + i * 4;
            pl = fmaf(v.x, sW2[d+0], pl); pl = fmaf(v.y, sW2[d+1], pl);
            pl = fmaf(v.z, sW2[d+2], pl); pl = fmaf(v.w, sW2[d+3], pl);
            pr = fmaf(v.x, sW3[d+0], pr); pr = fmaf(v.y, sW3[d+1], pr);
            pr = fmaf(v.z, sW3[d+2], pr); pr = fmaf(v.w, sW3[d+3], pr);
        }
        sPart[row * 8 + q]     = pl;
        sPart[row * 8 + 4 + q] = pr;
    }
    __syncthreads();
    if (tid < NN) {
        const float* p = &sPart[tid * 8];
        sLhs[tid] = (p[0] + p[1] + p[2] + p[3]) * w1;
        sRhs[tid] =  p[4] + p[5] + p[6] + p[7];
    }
    __syncthreads();

    // ---- phase 2: t = tanh(outer(lhs, rhs) + bs) ----
#pragma unroll
    for (int i = 0; i < 16; ++i) {
        int idx = tid + i * 256;
        int j = idx >> 6, k = idx & 63;
        sT[j * PITCH + k] = tanhf(sLhs[j] * sRhs[k] + bs[idx]);
    }
    __syncthreads();

    // ---- phase 3: s = Vs @ t via V_WMMA_F32_16X16X4_F32 ----
    const int lo = lane & 15;
    const int hi = lane >> 4;      // 0 or 1 (lane half)
    v8f acc[2];
#pragma unroll
    for (int tsel = 0; tsel < 2; ++tsel) {
        const int tile = wave + tsel * 8;        // 16 tiles, 8 waves
        const int mi = (tile >> 2) * 16;         // output row block
        const int nk = (tile &  3) * 16;         // output col block
        v8f c = {0.f,0.f,0.f,0.f,0.f,0.f,0.f,0.f};
#pragma unroll
        for (int j0 = 0; j0 < NN; j0 += 4) {     // K steps of 4
            const int kc = j0 + hi * 2;
            v2f a, bb;
            a.x  = sVs[(mi + lo) * PITCH + kc + 0];   // A 16x4: VGPR0=K0/K2, VGPR1=K1/K3
            a.y  = sVs[(mi + lo) * PITCH + kc + 1];
            bb.x = sT[(kc + 0) * PITCH + nk + lo];    // B 4x16: VGPR0 rows K0/K2, VGPR1 K1/K3
            bb.y = sT[(kc + 1) * PITCH + nk + lo];
            c = __builtin_amdgcn_wmma_f32_16x16x4_f32(
                    false, a, false, bb, (short)0, c, false, false);
        }
        acc[tsel] = c;
    }
    __syncthreads();               // all waves done reading sT
#pragma unroll
    for (int tsel = 0; tsel < 2; ++tsel) {
        const int tile = wave + tsel * 8;
        const int mi = (tile >> 2) * 16;
        const int nk = (tile &  3) * 16;
#pragma unroll
        for (int r = 0; r < 8; ++r)              // C/D: VGPR r -> M=r (+8 for hi lanes)
            sT[(mi + r + hi * 8) * PITCH + nk + lo] = acc[tsel][r];
    }
    __syncthreads();

    // ---- phase 4: softmax over all 4096 scores ----
    float m = -3.4e38f;
#pragma unroll
    for (int i = 0; i < 16; ++i) {
        int idx = tid + i * 256;
        m = fmaxf(m, sT[(idx >> 6) * PITCH + (idx & 63)]);
    }
#pragma unroll
    for (int off = 16; off > 0; off >>= 1)
        m = fmaxf(m, __shfl_down(m, off, 32));
    if (lane == 0) sRed[wave] = m;
    __syncthreads();
    float gmax = sRed[0];
#pragma unroll
    for (int w = 1; w < 8; ++w) gmax = fmaxf(gmax, sRed[w]);
    __syncthreads();               // done reading sRed before reuse

    float ssum = 0.f;
#pragma unroll
    for (int i = 0; i < 16; ++i) {
        int idx = tid + i * 256;
        float* p = &sT[(idx >> 6) * PITCH + (idx & 63)];
        float e = __expf(*p - gmax);
        *p = e;                    // each element owned by exactly one thread
        ssum += e;
    }
#pragma unroll
    for (int off = 16; off > 0; off >>= 1)
        ssum += __shfl_down(ssum, off, 32);
    if (lane == 0) sRed[wave] = ssum;
    __syncthreads();
    float gsum = 0.f;
#pragma unroll
    for (int w = 0; w < 8; ++w) gsum += sRed[w];
    const float inv = 1.0f / gsum;

    float* op = out + (size_t)b * (NN * NN);
#pragma unroll
    for (int i = 0; i < 16; ++i) {
        int idx = tid + i * 256;
        op[idx] = sT[(idx >> 6) * PITCH + (idx & 63)] * inv;
    }
}

extern "C" void kernel_launch(void* const* d_in, const int* in_sizes, int n_in,
                              void* d_out, int out_size, void* d_ws, size_t ws_size,
                              hipStream_t stream) {
    const float* x  = (const float*)d_in[0];
    const float* W1 = (const float*)d_in[1];
    const float* W2 = (const float*)d_in[2];
    const float* W3 = (const float*)d_in[3];
    const float* bs = (const float*)d_in[4];
    const float* Vs = (const float*)d_in[5];
    float* out = (float*)d_out;

    const int Bbatch = in_sizes[0] / (NN * DD);  // 4096
    dim3 grid(Bbatch), block(256);
    adj_attn_kernel<<<grid, block, 0, stream>>>(x, W1, W2, W3, bs, Vs, out);
}